// ActorGNN_71751723647616
// MI455X (gfx1250) — compile-verified
//
#include <hip/hip_runtime.h>
#include <hip/hip_bf16.h>
#include <math.h>

#define N_NODES 100000
#define N_EDGES 1600000
#define DIM     64

typedef __attribute__((ext_vector_type(2))) float v2f;
typedef __attribute__((ext_vector_type(8))) float v8f;

// ---------------- degree: deg[dst] += 1 ----------------
__global__ void degree_kernel(const long long* __restrict__ ei,
                              float* __restrict__ deg) {
    int e = blockIdx.x * blockDim.x + threadIdx.x;
    if (e < N_EDGES) {
        long long dst = ei[N_EDGES + e];
        atomicAdd(&deg[dst], 1.0f);
    }
}

// ---------------- deg -> 1/max(deg,1) in place ----------------
__global__ void deginv_kernel(float* __restrict__ deg) {
    int i = blockIdx.x * blockDim.x + threadIdx.x;
    if (i < N_NODES) deg[i] = 1.0f / fmaxf(deg[i], 1.0f);
}

// ---------------- scatter-add: agg[dst] += h[src] ----------------
// 4 threads per edge, each handles 16 consecutive features as 4x float4.
__global__ void scatter_kernel(const long long* __restrict__ ei,
                               const float* __restrict__ h,
                               float* __restrict__ agg) {
    int tid = blockIdx.x * blockDim.x + threadIdx.x;
    int e = tid >> 2;
    int q = tid & 3;
    if (e < N_EDGES) {
        long long src = ei[e];
        long long dst = ei[N_EDGES + e];
        const float4* hp = (const float4*)(h + (size_t)src * DIM + q * 16);
        float* ap = agg + (size_t)dst * DIM + q * 16;
#pragma unroll
        for (int j = 0; j < 4; ++j) {
            float4 v = hp[j];
            atomicAdd(ap + j * 4 + 0, v.x);
            atomicAdd(ap + j * 4 + 1, v.y);
            atomicAdd(ap + j * 4 + 2, v.z);
            atomicAdd(ap + j * 4 + 3, v.w);
        }
    }
}

// ---------------- fused dual-GEMM layer via V_WMMA_F32_16X16X4_F32 ----------
// out[N,64] = relu( h[N,64] @ Wself + (agg*deg_inv)[N,64] @ Wneigh + b )
// One wave per 16x16 output tile. 100000 rows = 6250 exact M tiles, 4 N tiles.
// f32 WMMA layouts (ISA 7.12.2):
//   A 16x4 : A[v][lane] = Amat[lane%16][kb + v + 2*(lane/16)]
//   B 4x16 : B[v][lane] = Bmat[kb + v + 2*(lane/16)][lane%16]
//   C 16x16: C[r][lane] = Cmat[r + 8*(lane/16)][lane%16]
__launch_bounds__(256)
__global__ void gemm_layer_kernel(const float* __restrict__ h,
                                  const float* __restrict__ agg,
                                  const float* __restrict__ deg_inv,
                                  const float* __restrict__ Wself,
                                  const float* __restrict__ Wneigh,
                                  const float* __restrict__ bias,
                                  float* __restrict__ out,
                                  int do_relu) {
    int wid  = blockIdx.x * (blockDim.x >> 5) + (threadIdx.x >> 5);
    int lane = threadIdx.x & 31;
    int tn = wid & 3;        // N tile (0..3)
    int tm = wid >> 2;       // M tile (0..6249)
    int m0 = tm * 16;
    int n0 = tn * 16;
    int rlo = lane & 15;     // A-row / B,C,D-column index
    int hi  = lane >> 4;     // half selector

    float dinv = deg_inv[m0 + rlo];
    const float* arow = h   + (size_t)(m0 + rlo) * DIM;
    const float* grow = agg + (size_t)(m0 + rlo) * DIM;

    v8f c = {0.f, 0.f, 0.f, 0.f, 0.f, 0.f, 0.f, 0.f};

#pragma unroll
    for (int kb = 0; kb < DIM; kb += 4) {
        int ka = kb + 2 * hi;
        // self-path: A = h tile, B = Wself tile
        v2f a0 = *(const v2f*)(arow + ka);
        v2f b0;
        b0.x = Wself[(ka    ) * DIM + n0 + rlo];
        b0.y = Wself[(ka + 1) * DIM + n0 + rlo];
        c = __builtin_amdgcn_wmma_f32_16x16x4_f32(false, a0, false, b0,
                                                  (short)0, c, false, false);
        // neighbor-path: A = (agg * deg_inv) tile, B = Wneigh tile
        v2f a1 = *(const v2f*)(grow + ka);
        a1.x *= dinv;
        a1.y *= dinv;
        v2f b1;
        b1.x = Wneigh[(ka    ) * DIM + n0 + rlo];
        b1.y = Wneigh[(ka + 1) * DIM + n0 + rlo];
        c = __builtin_amdgcn_wmma_f32_16x16x4_f32(false, a1, false, b1,
                                                  (short)0, c, false, false);
    }

    float bv = bias[n0 + rlo];
#pragma unroll
    for (int r = 0; r < 8; ++r) {
        int m = m0 + r + 8 * hi;
        float v = c[r] + bv;
        if (do_relu) v = fmaxf(v, 0.0f);
        out[(size_t)m * DIM + n0 + rlo] = v;
    }
}

// ---------------- final 64->1 layer + sigmoid ----------------
__global__ void final_kernel(const float* __restrict__ h,
                             const float* __restrict__ agg,
                             const float* __restrict__ deg_inv,
                             const float* __restrict__ Ws,
                             const float* __restrict__ Wn,
                             const float* __restrict__ b,
                             float* __restrict__ out) {
    int i = blockIdx.x * blockDim.x + threadIdx.x;
    if (i >= N_NODES) return;
    const float* hp = h   + (size_t)i * DIM;
    const float* ap = agg + (size_t)i * DIM;
    float di = deg_inv[i];
    float acc = b[0];
#pragma unroll 8
    for (int k = 0; k < DIM; ++k)
        acc += hp[k] * Ws[k] + di * ap[k] * Wn[k];
    out[i] = 1.0f / (1.0f + expf(-acc));
}

extern "C" void kernel_launch(void* const* d_in, const int* in_sizes, int n_in,
                              void* d_out, int out_size, void* d_ws, size_t ws_size,
                              hipStream_t stream) {
    (void)in_sizes; (void)n_in; (void)out_size; (void)ws_size;
    const float*     x   = (const float*)d_in[0];
    const long long* ei  = (const long long*)d_in[1];   // int64 edge_index [2,E]
    const float* Ws1 = (const float*)d_in[2];
    const float* Wn1 = (const float*)d_in[3];
    const float* b1  = (const float*)d_in[4];
    const float* Ws2 = (const float*)d_in[5];
    const float* Wn2 = (const float*)d_in[6];
    const float* b2  = (const float*)d_in[7];
    const float* Ws3 = (const float*)d_in[8];
    const float* Wn3 = (const float*)d_in[9];
    const float* b3  = (const float*)d_in[10];
    const float* Ws4 = (const float*)d_in[11];
    const float* Wn4 = (const float*)d_in[12];
    const float* b4  = (const float*)d_in[13];
    float* out = (float*)d_out;

    // workspace layout (floats): deg/deg_inv [N] | agg [N*64] | hA [N*64] | hB [N*64]
    float* deg = (float*)d_ws;
    float* agg = deg + N_NODES;
    float* hA  = agg + (size_t)N_NODES * DIM;
    float* hB  = hA  + (size_t)N_NODES * DIM;

    const int eblk = (N_EDGES + 255) / 256;
    const int sblk = (N_EDGES * 4 + 255) / 256;
    const int nblk = (N_NODES + 255) / 256;
    const int gblk = (N_NODES / 16) * 4 / 8;   // 6250 M-tiles * 4 N-tiles / 8 waves per block
    const size_t aggBytes = (size_t)N_NODES * DIM * sizeof(float);

    // degrees -> deg_inv (in place)
    hipMemsetAsync(deg, 0, N_NODES * sizeof(float), stream);
    degree_kernel<<<eblk, 256, 0, stream>>>(ei, deg);
    deginv_kernel<<<nblk, 256, 0, stream>>>(deg);

    // layer 1: x -> hA
    hipMemsetAsync(agg, 0, aggBytes, stream);
    scatter_kernel<<<sblk, 256, 0, stream>>>(ei, x, agg);
    gemm_layer_kernel<<<gblk, 256, 0, stream>>>(x, agg, deg, Ws1, Wn1, b1, hA, 1);

    // layer 2: hA -> hB
    hipMemsetAsync(agg, 0, aggBytes, stream);
    scatter_kernel<<<sblk, 256, 0, stream>>>(ei, hA, agg);
    gemm_layer_kernel<<<gblk, 256, 0, stream>>>(hA, agg, deg, Ws2, Wn2, b2, hB, 1);

    // layer 3: hB -> hA
    hipMemsetAsync(agg, 0, aggBytes, stream);
    scatter_kernel<<<sblk, 256, 0, stream>>>(ei, hB, agg);
    gemm_layer_kernel<<<gblk, 256, 0, stream>>>(hB, agg, deg, Ws3, Wn3, b3, hA, 1);

    // layer 4: hA -> out (64 -> 1, sigmoid)
    hipMemsetAsync(agg, 0, aggBytes, stream);
    scatter_kernel<<<sblk, 256, 0, stream>>>(ei, hA, agg);
    final_kernel<<<nblk, 256, 0, stream>>>(hA, agg, deg, Ws4, Wn4, b4, out);
}